// PostProcess_6382321402279
// MI455X (gfx1250) — compile-verified
//
#include <hip/hip_runtime.h>
#include <hip/hip_bf16.h>
#include <stdint.h>

// ---------------- problem constants (from reference) ----------------
#define BN        340704        // B*N = 32*10647
#define CSTRIDE   85            // 4 box + 1 conf + 80 cls
#define NCLS      80
#define SCORE_THR 0.3f
#define IOU_THR   0.5f
#define AREA_THR  0.005f
#define MAX_OUT   10
#define INV_SIZE  (1.0f / 416.0f)
#define NEG_INF   (-__builtin_inff())

// Wave tile: 32 boxes * 85 floats * 4B = 10,880 bytes = 680 x 16B chunks
#define WAVE_TILE_BYTES  (32 * CSTRIDE * 4)
#define WAVE_CHUNKS      (WAVE_TILE_BYTES / 16)     // 680
#define FULL_ROUNDS      (WAVE_CHUNKS / 32)         // 21
#define TAIL_CHUNKS      (WAVE_CHUNKS % 32)         // 8

// ---------------------------------------------------------------------------
// Stage 1: stream y_pred through LDS via CDNA5 async global->LDS (ASYNCcnt),
// compute score/class/normalized boxes per box.  256 thr = 8 wave32 / block,
// 32 boxes per wave, per-wave LDS tile (no cross-wave sync needed).
// ---------------------------------------------------------------------------
__global__ __launch_bounds__(256) void k_prep(const float* __restrict__ y,
                                              float* __restrict__ s_active,
                                              float* __restrict__ boxes_n,
                                              float* __restrict__ scores,
                                              int*   __restrict__ classes,
                                              int*   __restrict__ sel)
{
    __shared__ __align__(16) float tile[256 * CSTRIDE];   // 87,040 B

    const int lane = threadIdx.x & 31;
    const int wv   = threadIdx.x >> 5;

    const long long blockBox = (long long)blockIdx.x * 256;
    const long long waveBox  = blockBox + (long long)wv * 32;
    if (waveBox >= BN) return;            // whole-wave uniform exit (BN % 32 == 0)

    // Block-uniform 64-bit global base (stays in SGPRs: kernarg + blockIdx).
    const uint64_t gbase = (uint64_t)(uintptr_t)y + (uint64_t)blockBox * (CSTRIDE * 4);
    // LDS byte address of this block's tile (low 32 bits of generic shared ptr).
    const uint32_t ldsBase = (uint32_t)(uintptr_t)(&tile[0]);
    const uint32_t waveOff = (uint32_t)wv * WAVE_TILE_BYTES;

    // Async copy 680 x 16B chunks (wave moves 512B per instruction).
#pragma unroll
    for (int r = 0; r < FULL_ROUNDS; ++r) {
        uint32_t voff  = waveOff + (uint32_t)(lane + r * 32) * 16u;
        uint32_t laddr = ldsBase + voff;
        asm volatile("global_load_async_to_lds_b128 %0, %1, %2"
                     :: "v"(laddr), "v"(voff), "s"(gbase) : "memory");
    }
    if (lane < TAIL_CHUNKS) {             // predicated tail (partial EXEC is legal)
        uint32_t voff  = waveOff + (uint32_t)(FULL_ROUNDS * 32 + lane) * 16u;
        uint32_t laddr = ldsBase + voff;
        asm volatile("global_load_async_to_lds_b128 %0, %1, %2"
                     :: "v"(laddr), "v"(voff), "s"(gbase) : "memory");
    }
    asm volatile("s_wait_asynccnt 0" ::: "memory");

    // Each lane owns one row.  Row stride 85 floats: 85 mod 64 banks = 21,
    // gcd(21,64)=1 -> conflict-free LDS reads.
    const float* row = &tile[(wv * 32 + lane) * CSTRIDE];
    const long long gi = waveBox + lane;

    float x1 = row[0], y1 = row[1], x2 = row[2], y2 = row[3];
    float conf = row[4];

    float best = row[5];
    int   bidx = 0;
#pragma unroll 8
    for (int k = 1; k < NCLS; ++k) {
        float v = row[5 + k];
        if (v > best) { best = v; bidx = k; }   // strict '>' == first-occurrence argmax
    }
    float score = conf * best;

    float4 bn = make_float4(x1 * INV_SIZE, y1 * INV_SIZE, x2 * INV_SIZE, y2 * INV_SIZE);
    ((float4*)boxes_n)[gi] = bn;                 // 16B-aligned store
    scores[gi]  = score;
    classes[gi] = bidx;
    sel[gi]     = 0;                             // self-initialize workspace
    s_active[gi] = (score > SCORE_THR) ? score : NEG_INF;
}

// ---------------------------------------------------------------------------
// Stage 2: sequential greedy NMS, 10 rounds, one 1024-thread workgroup.
// Each thread owns indices {tid, tid+1024, ...} so it only ever re-reads its
// OWN global writes to s_active (same-thread store->load order is guaranteed).
// Score array (1.36 MB) lives in L2 (192 MB) across rounds.
// ---------------------------------------------------------------------------
__global__ __launch_bounds__(1024) void k_nms(const float* __restrict__ boxes_n,
                                              float* __restrict__ s_active,
                                              int*   __restrict__ sel)
{
    __shared__ float rv[1024];
    __shared__ int   ri[1024];
    __shared__ float bbox[4];
    __shared__ int   okf;

    const int tid = threadIdx.x;

    for (int it = 0; it < MAX_OUT; ++it) {
        // local argmax over this thread's slice (first-index tie-break)
        float bv = NEG_INF;
        int   bi = 0x7fffffff;
        for (int i = tid; i < BN; i += 1024) {
            float v = s_active[i];
            if (v > bv || (v == bv && i < bi)) { bv = v; bi = i; }
        }
        rv[tid] = bv; ri[tid] = bi;
        __syncthreads();

        // LDS tree reduce to global argmax
        for (int s = 512; s > 0; s >>= 1) {
            if (tid < s) {
                float ov = rv[tid + s]; int oi = ri[tid + s];
                if (ov > rv[tid] || (ov == rv[tid] && oi < ri[tid])) {
                    rv[tid] = ov; ri[tid] = oi;
                }
            }
            __syncthreads();
        }

        if (tid == 0) {
            int ok = (rv[0] > NEG_INF) ? 1 : 0;   // ok = s[idx] > -inf
            okf = ok;
            if (ok) {
                int b = ri[0];
                sel[b] = 1;
                float4 bb = ((const float4*)boxes_n)[b];
                bbox[0] = bb.x; bbox[1] = bb.y; bbox[2] = bb.z; bbox[3] = bb.w;
            }
        }
        __syncthreads();

        if (okf) {
            const float bx1 = bbox[0], by1 = bbox[1], bx2 = bbox[2], by2 = bbox[3];
            const float barea = (bx2 - bx1) * (by2 - by1);
            for (int i = tid; i < BN; i += 1024) {
                float v = s_active[i];
                if (v == NEG_INF) continue;        // already dead, can't revive
                float4 c = ((const float4*)boxes_n)[i];
                float ix1 = fmaxf(c.x, bx1);
                float iy1 = fmaxf(c.y, by1);
                float ix2 = fminf(c.z, bx2);
                float iy2 = fminf(c.w, by2);
                float inter = fmaxf(ix2 - ix1, 0.0f) * fmaxf(iy2 - iy1, 0.0f);
                float area  = (c.z - c.x) * (c.w - c.y);
                float iou   = inter / (area + barea - inter);
                if (iou > IOU_THR) s_active[i] = NEG_INF;   // suppress (kills winner too)
            }
        }
        __syncthreads();   // protect rv/ri reuse next round
    }
}

// ---------------------------------------------------------------------------
// Stage 3: masked write-out.  d_out = [boxes(BN*4) | scores(BN) | classes(BN)]
// ---------------------------------------------------------------------------
__global__ __launch_bounds__(256) void k_out(const float* __restrict__ boxes_n,
                                             const float* __restrict__ scores,
                                             const int*   __restrict__ classes,
                                             const int*   __restrict__ sel,
                                             float* __restrict__ out)
{
    int i = blockIdx.x * 256 + threadIdx.x;
    if (i >= BN) return;

    float4 b = ((const float4*)boxes_n)[i];
    float area = (b.z - b.x) * (b.w - b.y);
    bool  m = (sel[i] != 0) && (area > AREA_THR);
    float f = m ? 1.0f : 0.0f;

    float4 ob = make_float4(b.x * f, b.y * f, b.z * f, b.w * f);
    ((float4*)out)[i]   = ob;
    out[4 * BN + i]     = scores[i] * f;
    out[5 * BN + i]     = m ? (float)classes[i] : -1.0f;
}

// ---------------------------------------------------------------------------
extern "C" void kernel_launch(void* const* d_in, const int* in_sizes, int n_in,
                              void* d_out, int out_size, void* d_ws, size_t ws_size,
                              hipStream_t stream)
{
    (void)in_sizes; (void)n_in; (void)out_size; (void)ws_size;

    const float* y = (const float*)d_in[0];

    // workspace carve-up: 8*BN floats (~10.9 MB)
    float* ws       = (float*)d_ws;
    float* s_active = ws;                 // BN
    float* boxes_n  = ws + (size_t)BN;    // BN*4 (16B aligned: BN*4B offset is mult of 16)
    float* scores   = ws + (size_t)5 * BN;
    int*   classes  = (int*)(ws + (size_t)6 * BN);
    int*   sel      = (int*)(ws + (size_t)7 * BN);

    const int prepBlocks = (BN + 255) / 256;   // 1331 (every wave full or empty)
    k_prep<<<prepBlocks, 256, 0, stream>>>(y, s_active, boxes_n, scores, classes, sel);
    k_nms <<<1, 1024, 0, stream>>>(boxes_n, s_active, sel);
    k_out <<<(BN + 255) / 256, 256, 0, stream>>>(boxes_n, scores, classes, sel, (float*)d_out);
}